// IDWT_2D_38465727103452
// MI455X (gfx1250) — compile-verified
//
#include <hip/hip_runtime.h>

// Inverse 2x2 Haar DWT, memory-bound streaming kernel for MI455X (gfx1250).
// Input : x[8][256][128][128] f32  (channels = 4 subbands x 64)
// Output: out[8][64][256][256] f32
//
// Per input pixel (ll,lh,hl,hh):
//   a=ll+lh; b=ll-lh; c=hl+hh; d=hl-hh;
//   out[2h  ][2w]=a+c; out[2h  ][2w+1]=a-c;
//   out[2h+1][2w]=b+d; out[2h+1][2w+1]=b-d;
//
// One thread handles 2 consecutive w-pixels:
//   4 x b64 NT loads (dense 256B/wave/plane), 2 x b128 NT stores (dense 512B/wave).

typedef float v2f __attribute__((ext_vector_type(2)));
typedef float v4f __attribute__((ext_vector_type(4)));

__device__ __forceinline__ v2f ld_nt2(const float* __restrict__ p) {
    return __builtin_nontemporal_load((const v2f*)p);
}
__device__ __forceinline__ void st_nt4(float* __restrict__ p, v4f v) {
    __builtin_nontemporal_store(v, (v4f*)p);
}

__global__ __launch_bounds__(256) void idwt_haar_nt_kernel(
    const float* __restrict__ x, float* __restrict__ out)
{
    // Fixed problem geometry from the reference.
    constexpr int C   = 64;       // output channels
    constexpr int H   = 128;      // input height
    constexpr int W   = 128;      // input width
    constexpr int W2  = W / 2;    // pixels-per-thread granularity: 2 -> 64 threads/row
    constexpr size_t PLANE  = (size_t)H * W;           // 16384   (one channel plane)
    constexpr size_t SBAND  = (size_t)C * PLANE;       // 1048576 (subband stride in elems)
    constexpr size_t OPITCH = (size_t)2 * W;           // 256     (output row pitch)

    const unsigned tid = blockIdx.x * blockDim.x + threadIdx.x;
    // tid -> (b, c, h, w2);  totals: b:8, c:64, h:128, w2:64
    const unsigned w2 = tid & (W2 - 1);
    const unsigned h  = (tid >> 6) & (H - 1);
    const unsigned c  = (tid >> 13) & (C - 1);
    const unsigned b  = tid >> 19;

    // Input: channel (s*64 + c) of batch b, subband s in {0,1,2,3}.
    const float* __restrict__ pin =
        x + (((size_t)b * 256u + c) * H + h) * W + 2u * w2;

    const v2f ll = ld_nt2(pin + 0 * SBAND);
    const v2f lh = ld_nt2(pin + 1 * SBAND);
    const v2f hl = ld_nt2(pin + 2 * SBAND);
    const v2f hh = ld_nt2(pin + 3 * SBAND);

    const v2f a = ll + lh;
    const v2f s = ll - lh;   // "b" term
    const v2f t = hl + hh;   // "c" term
    const v2f d = hl - hh;

    // Interleave in registers: row0 = (tl0, tr0, tl1, tr1), row1 = (bl0, br0, bl1, br1)
    v4f row0, row1;
    row0.x = a.x + t.x;  row0.y = a.x - t.x;
    row0.z = a.y + t.y;  row0.w = a.y - t.y;
    row1.x = s.x + d.x;  row1.y = s.x - d.x;
    row1.z = s.y + d.y;  row1.w = s.y - d.y;

    float* __restrict__ po =
        out + (((size_t)b * C + c) * (2 * H) + 2u * h) * OPITCH + 4u * w2;

    st_nt4(po,          row0);
    st_nt4(po + OPITCH, row1);
}

extern "C" void kernel_launch(void* const* d_in, const int* in_sizes, int n_in,
                              void* d_out, int out_size, void* d_ws, size_t ws_size,
                              hipStream_t stream) {
    (void)in_sizes; (void)n_in; (void)d_ws; (void)ws_size; (void)out_size;
    const float* x   = (const float*)d_in[0];
    float*       out = (float*)d_out;

    // Threads = 8 * 64 * 128 * 64 = 4,194,304 ; block = 256 (8 waves, wave32)
    constexpr unsigned total  = 8u * 64u * 128u * 64u;
    constexpr unsigned block  = 256u;
    constexpr unsigned grid   = total / block;   // 16384

    idwt_haar_nt_kernel<<<grid, block, 0, stream>>>(x, out);
}